// ConditionalRandomField_23837068493071
// MI455X (gfx1250) — compile-verified
//
#include <hip/hip_runtime.h>

typedef __attribute__((ext_vector_type(2))) float v2f;
typedef __attribute__((ext_vector_type(8))) float v8f;

#define CRF_B 128
#define CRF_T 4096
#define CRF_K 64
#define BPB   16     // batches per block (M tile = 16)
#define LSTR  68     // padded LDS row stride in floats (conflict-free frag loads)

#define LOG2E 1.4426950408889634f
#define LN2   0.6931471805599453f

// Raw hardware transcendentals (v_exp_f32 = exp2, v_log_f32 = log2).
// Safe here: exp2 args bounded, log2 args >= ~0.3 (never denormal).
__device__ __forceinline__ float fast_exp2(float x) { return __builtin_amdgcn_exp2f(x); }
__device__ __forceinline__ float fast_log2(float x) { return __builtin_amdgcn_logf(x); }

// ---------------------------------------------------------------------------
// Forward algorithm (denominator).  One block = 16 batches, 4 waves.
// logsumexp_i(alpha_i + trans_ij) = c + log(sum_i exp(alpha_i - c) * E_ij),
// E = exp(transitions) constant -> its WMMA B-fragments live in registers.
// Offset c = alpha[b][0] (inter-state spread bounded by CRF contraction).
// Per step: Phase A builds u = exp(alpha - c) in LDS; Phase B runs a dual-
// chain f32 WMMA GEMM U(16x64)xE(64x64) tile + log + emission (prefetch d=2).
// Mask handling is a pure FMA blend (mask is 0/1) so no exec-branches can be
// formed in the 4096-iteration critical path.
// ---------------------------------------------------------------------------
__global__ __launch_bounds__(128) void crf_forward(
    const float* __restrict__ logits,
    const float* __restrict__ transitions,
    const float* __restrict__ start_transitions,
    const float* __restrict__ end_transitions,
    const int*   __restrict__ mask,
    float* __restrict__ den)
{
  __shared__ __align__(16) float alphaBuf[BPB * LSTR];
  __shared__ __align__(16) float uBuf[BPB * LSTR];
  __shared__ __align__(16) float mArr[BPB];
  __shared__ __align__(16) float mskArr[BPB];

  const int tid  = threadIdx.x;
  const int wave = tid >> 5;
  const int lane = tid & 31;
  const int half = lane >> 4;
  const int ln   = lane & 15;
  const int b0   = blockIdx.x * BPB;

  // B-fragments of E = exp(transitions) for this wave's 16 output states.
  // 16x16x4 f32 B frag: VGPR0 = row (4kk+2*half), VGPR1 = row +1, col = ln.
  v2f bfrag[16];
  {
    const int n = wave * 16 + ln;
#pragma unroll
    for (int kk = 0; kk < 16; ++kk) {
      const int kr = 4 * kk + 2 * half;
      bfrag[kk].x = fast_exp2(transitions[kr * CRF_K + n] * LOG2E);
      bfrag[kk].y = fast_exp2(transitions[(kr + 1) * CRF_K + n] * LOG2E);
    }
  }

  // alpha0 = start_transitions + logits[:,0,:]  (LDS copy, Phase A layout)
  {
    const int b = tid >> 3, sub = tid & 7, j0 = sub * 8;
#pragma unroll
    for (int i = 0; i < 8; ++i) {
      const int jj = j0 + i;
      alphaBuf[b * LSTR + jj] =
          start_transitions[jj] + logits[((size_t)(b0 + b) * CRF_T) * CRF_K + jj];
    }
  }

  const int j = wave * 16 + ln;   // this lane's output state (column)

  // Register copy of this thread's owned alpha elements (rows r+8*half).
  float aReg[8];
#pragma unroll
  for (int r = 0; r < 8; ++r) {
    const int bl = r + 8 * half;
    aReg[r] = start_transitions[j] + logits[((size_t)(b0 + bl) * CRF_T) * CRF_K + j];
  }

  // Emission double buffer, prefetch distance 2.
  float em_cur[8], em_nxt[8];
#pragma unroll
  for (int r = 0; r < 8; ++r) {
    const int bl = r + 8 * half;
    em_cur[r] = logits[((size_t)(b0 + bl) * CRF_T + 1) * CRF_K + j];
    em_nxt[r] = logits[((size_t)(b0 + bl) * CRF_T + 2) * CRF_K + j];
  }
  __syncthreads();

#pragma unroll 2
  for (int t = 1; t < CRF_T; ++t) {
    // ---- Phase A: u = exp(alpha - alpha[:,0]) into uBuf ------------------
    {
      const int b = tid >> 3, sub = tid & 7, j0 = sub * 8;
      const float c = alphaBuf[b * LSTR];          // broadcast read, offset
      float v[8];
#pragma unroll
      for (int i = 0; i < 8; ++i) v[i] = alphaBuf[b * LSTR + j0 + i];
#pragma unroll
      for (int i = 0; i < 8; ++i)
        uBuf[b * LSTR + j0 + i] = fast_exp2((v[i] - c) * LOG2E);
      if (sub == 0) {
        mArr[b]   = c;
        mskArr[b] = (float)mask[(size_t)(b0 + b) * CRF_T + t];
      }
    }
    __syncthreads();

    // ---- Phase B ---------------------------------------------------------
    {
      // Per-row constants: two b128 LDS reads per array.
      const float4 c01 = ((const float4*)(&mArr[8 * half]))[0];
      const float4 c23 = ((const float4*)(&mArr[8 * half]))[1];
      const float4 m01 = ((const float4*)(&mskArr[8 * half]))[0];
      const float4 m23 = ((const float4*)(&mskArr[8 * half]))[1];
      const float crow[8] = {c01.x, c01.y, c01.z, c01.w, c23.x, c23.y, c23.z, c23.w};
      const float mrow[8] = {m01.x, m01.y, m01.z, m01.w, m23.x, m23.y, m23.z, m23.w};

      // Prefetch emissions for t+2 while the GEMM runs (distance 2).
      const int tn = (t + 2 < CRF_T) ? (t + 2) : (CRF_T - 1);
      float em_new[8];
#pragma unroll
      for (int r = 0; r < 8; ++r) {
        const int bl = r + 8 * half;
        em_new[r] = logits[((size_t)(b0 + bl) * CRF_T + tn) * CRF_K + j];
      }

      // Load all 16 A-fragments (clause of ds_load_2addr_b64).
      v2f afrag[16];
#pragma unroll
      for (int kk = 0; kk < 16; ++kk) {
        const float* up = &uBuf[ln * LSTR + 4 * kk + 2 * half];
        afrag[kk].x = up[0];
        afrag[kk].y = up[1];
      }

      // Dual independent accumulator chains over K = 64 (8 kk-pairs).
      v8f acc0 = {0.f, 0.f, 0.f, 0.f, 0.f, 0.f, 0.f, 0.f};
      v8f acc1 = {0.f, 0.f, 0.f, 0.f, 0.f, 0.f, 0.f, 0.f};
#pragma unroll
      for (int p = 0; p < 8; ++p) {
        acc0 = __builtin_amdgcn_wmma_f32_16x16x4_f32(
            false, afrag[2 * p], false, bfrag[2 * p], (short)0, acc0, false, false);
        acc1 = __builtin_amdgcn_wmma_f32_16x16x4_f32(
            false, afrag[2 * p + 1], false, bfrag[2 * p + 1], (short)0, acc1, false, false);
      }

      // alpha'[b][j] = c_b + ln(S) + emit ; mask blend is a pure FMA
      // (mrow is exactly 0.0 or 1.0) -> branchless by construction.
#pragma unroll
      for (int r = 0; r < 8; ++r) {
        const int   bl   = r + 8 * half;
        const float S    = acc0[r] + acc1[r];
        const float cand = fmaf(LN2, fast_log2(S), crow[r]) + em_cur[r];
        aReg[r] = fmaf(mrow[r], cand - aReg[r], aReg[r]);
        alphaBuf[bl * LSTR + j] = aReg[r];     // unconditional store
        em_cur[r] = em_nxt[r];
        em_nxt[r] = em_new[r];
      }
    }
    __syncthreads();
  }

  // ---- final: den_b = logsumexp_j(alpha[b][j] + end_transitions[j]) ------
  {
    const int b = tid >> 3, sub = tid & 7, j0 = sub * 8;
    float v[8];
#pragma unroll
    for (int i = 0; i < 8; ++i)
      v[i] = alphaBuf[b * LSTR + j0 + i] + end_transitions[j0 + i];
    float mx = v[0];
#pragma unroll
    for (int i = 1; i < 8; ++i) mx = fmaxf(mx, v[i]);
    mx = fmaxf(mx, __shfl_xor(mx, 1, 8));
    mx = fmaxf(mx, __shfl_xor(mx, 2, 8));
    mx = fmaxf(mx, __shfl_xor(mx, 4, 8));
    float s = 0.f;
#pragma unroll
    for (int i = 0; i < 8; ++i) s += fast_exp2((v[i] - mx) * LOG2E);
    s += __shfl_xor(s, 1, 8);
    s += __shfl_xor(s, 2, 8);
    s += __shfl_xor(s, 4, 8);
    if (sub == 0) den[b0 + b] = fmaf(LN2, fast_log2(s), mx);
  }
}

// ---------------------------------------------------------------------------
// Joint (numerator) score: one block per batch, gathers along the tag path.
// ---------------------------------------------------------------------------
__global__ __launch_bounds__(256) void crf_joint(
    const float* __restrict__ logits,
    const float* __restrict__ transitions,
    const float* __restrict__ start_transitions,
    const float* __restrict__ end_transitions,
    const long long* __restrict__ tags,
    const int* __restrict__ mask,
    float* __restrict__ num)
{
  __shared__ float s_em[8], s_tr[8];
  __shared__ int   s_mc[8];
  const int b   = blockIdx.x;
  const int tid = threadIdx.x;
  float em = 0.f, tr = 0.f;
  int   mc = 0;
  for (int t = tid; t < CRF_T; t += 256) {
    const int tg = (int)tags[(size_t)b * CRF_T + t];
    const int mk = mask[(size_t)b * CRF_T + t];
    mc += mk;
    if (t < CRF_T - 1) {
      em += logits[((size_t)b * CRF_T + t) * CRF_K + tg] * (float)mk;
      const int tg2 = (int)tags[(size_t)b * CRF_T + t + 1];
      const int mk2 = mask[(size_t)b * CRF_T + t + 1];
      tr += transitions[tg * CRF_K + tg2] * (float)mk2;
    }
  }
  for (int m = 16; m > 0; m >>= 1) {
    em += __shfl_xor(em, m, 32);
    tr += __shfl_xor(tr, m, 32);
    mc += __shfl_xor(mc, m, 32);
  }
  if ((tid & 31) == 0) { s_em[tid >> 5] = em; s_tr[tid >> 5] = tr; s_mc[tid >> 5] = mc; }
  __syncthreads();
  if (tid == 0) {
    float EM = 0.f, TR = 0.f; int MC = 0;
    for (int i = 0; i < 8; ++i) { EM += s_em[i]; TR += s_tr[i]; MC += s_mc[i]; }
    const int   last_idx = MC - 1;
    const int   tag0     = (int)tags[(size_t)b * CRF_T];
    const int   last_tag = (int)tags[(size_t)b * CRF_T + last_idx];
    const float last_in  = logits[((size_t)b * CRF_T + (CRF_T - 1)) * CRF_K + last_tag];
    const float fm_last  = (float)mask[(size_t)b * CRF_T + CRF_T - 1];
    num[b] = start_transitions[tag0] + EM + TR + end_transitions[last_tag] + last_in * fm_last;
  }
}

// ---------------------------------------------------------------------------
// out[0] = sum_b (num[b] - den[b])
// ---------------------------------------------------------------------------
__global__ __launch_bounds__(128) void crf_final(
    const float* __restrict__ num, const float* __restrict__ den,
    float* __restrict__ out)
{
  __shared__ float s[4];
  const int tid = threadIdx.x;
  float v = num[tid] - den[tid];
  for (int m = 16; m > 0; m >>= 1) v += __shfl_xor(v, m, 32);
  if ((tid & 31) == 0) s[tid >> 5] = v;
  __syncthreads();
  if (tid == 0) out[0] = s[0] + s[1] + s[2] + s[3];
}

extern "C" void kernel_launch(void* const* d_in, const int* in_sizes, int n_in,
                              void* d_out, int out_size, void* d_ws, size_t ws_size,
                              hipStream_t stream) {
  const float*     logits            = (const float*)d_in[0];
  const float*     transitions       = (const float*)d_in[1];
  const float*     start_transitions = (const float*)d_in[2];
  const float*     end_transitions   = (const float*)d_in[3];
  const long long* tags              = (const long long*)d_in[4];
  const int*       maskp             = (const int*)d_in[5];

  float* num = (float*)d_ws;
  float* den = num + CRF_B;

  crf_joint<<<CRF_B, 256, 0, stream>>>(logits, transitions, start_transitions,
                                       end_transitions, tags, maskp, num);
  crf_forward<<<CRF_B / BPB, 128, 0, stream>>>(logits, transitions,
                                               start_transitions,
                                               end_transitions, maskp, den);
  crf_final<<<1, 128, 0, stream>>>(num, den, (float*)d_out);
}